// CTRGC_75823352644079
// MI455X (gfx1250) — compile-verified
//
#include <hip/hip_runtime.h>
#include <cstddef>

typedef float v2f __attribute__((ext_vector_type(2)));
typedef float v8f __attribute__((ext_vector_type(8)));

#define EPSF   1e-5f
#define CIN    64
#define TDIM   256
#define VV     18
#define KSUB   3
#define CO     64            // Cs = output channels
#define KC     192           // KSUB*CIN contraction length
#define TCHUNK 8
#define ROWS   (TCHUNK*VV)   // 144 = 9 * 16
#define ZSTR   68            // z LDS row stride (pad: conflict-free b64 frag loads)
#define WSTR   66            // Wf LDS row stride
#define OSTR   145           // out-staging stride
#define THREADS 512

// workspace float offsets
#define WF_OFF 0             // Wf[o][kc] : 64*192
#define BF_OFF 12288         // bf[o]     : 64
#define MM_OFF 12352         // M[k][w][u]: 3*18*18 = 972

// ---------------- setup: fold BNs into weights/bias, fold A+PA ----------------
__global__ void ctrgc_setup(const float* __restrict__ A,  const float* __restrict__ PA,
                            const float* __restrict__ Wta,const float* __restrict__ bta,
                            const float* __restrict__ g_ta,const float* __restrict__ b_ta,
                            const float* __restrict__ m_ta,const float* __restrict__ v_ta,
                            const float* __restrict__ g_bn,const float* __restrict__ b_bn,
                            const float* __restrict__ m_bn,const float* __restrict__ v_bn,
                            float* __restrict__ ws) {
    int tid = threadIdx.x;
    for (int i = tid; i < CO * KC; i += blockDim.x) {
        int o  = i / KC;
        int kc = i - o * KC;
        int k  = kc >> 6, c = kc & 63;
        float sk  = g_ta[k*64 + o] * rsqrtf(v_ta[k*64 + o] + EPSF);
        float sbn = g_bn[o]        * rsqrtf(v_bn[o]        + EPSF);
        ws[WF_OFF + i] = sbn * sk * Wta[k*4096 + o*64 + c];
    }
    for (int o = tid; o < CO; o += blockDim.x) {
        float sbn = g_bn[o] * rsqrtf(v_bn[o] + EPSF);
        float acc = 0.f;
        for (int k = 0; k < KSUB; ++k) {
            float sk = g_ta[k*64 + o] * rsqrtf(v_ta[k*64 + o] + EPSF);
            acc += sk * (bta[k*64 + o] - m_ta[k*64 + o]) + b_ta[k*64 + o];
        }
        ws[BF_OFF + o] = sbn * (acc - m_bn[o]) + b_bn[o];
    }
    for (int i = tid; i < KSUB*VV*VV; i += blockDim.x)
        ws[MM_OFF + i] = A[i] + PA[i];
}

// ---------------- main fused kernel ----------------
__global__ __launch_bounds__(THREADS)
void ctrgc_main(const float* __restrict__ x, const float* __restrict__ ws,
                float* __restrict__ out) {
    __shared__ __align__(16) float zbuf[ROWS * ZSTR];   // 9792 f (also reused as out-stage)
    __shared__ __align__(16) float wfs[CO * WSTR];      // 4224 f
    __shared__ __align__(16) float ms[KSUB * VV * VV];  // 972 f
    __shared__ __align__(16) float bfs[CO];

    const int tid  = threadIdx.x;
    const int wave = tid >> 5;
    const int lane = tid & 31;
    const int half = lane >> 4;
    const int l16  = lane & 15;

    const int n  = blockIdx.x >> 5;            // 128 n's
    const int t0 = (blockIdx.x & 31) * TCHUNK; // 32 t-chunks

    // prologue: small constants to LDS
    for (int i = tid; i < KSUB*VV*VV; i += THREADS) ms[i] = ws[MM_OFF + i];
    if (tid < CO) bfs[tid] = ws[BF_OFF + tid];

    // z-phase assignment: one (c, tt) per thread
    const int zc  = tid >> 3;   // 0..63 channel
    const int ztt = tid & 7;    // 0..7 t within chunk
    const float* xrow = x + ((size_t)(n*CIN + zc) * TDIM + (t0 + ztt)) * VV;

    __syncthreads();

    // accumulators: tiles wave, wave+16, wave+32 (36 tiles = 9 M x 4 N)
    v8f acc[3];
#pragma unroll
    for (int ti = 0; ti < 3; ++ti) {
        int tile = wave + 16*ti;
        if (tile < 36) {
            float b = bfs[(tile & 3)*16 + l16];
            v8f bi = {b, b, b, b, b, b, b, b};
            acc[ti] = bi;
        }
    }

    for (int k = 0; k < KSUB; ++k) {
        __syncthreads();   // previous chunk fully consumed

        // stage Wf chunk [o][64] for this k
        for (int i = tid; i < CO * 64; i += THREADS) {
            int o = i >> 6, j = i & 63;
            wfs[o*WSTR + j] = ws[WF_OFF + o*KC + k*64 + j];
        }

        // V-mix: z[row=(tt,v)][c] = sum_w x[n,c,t,w] * M_k[w,v]
        float xr[VV];
#pragma unroll
        for (int w = 0; w < VV; w += 2) {
            v2f t2 = *(const v2f*)(xrow + w);
            xr[w] = t2.x; xr[w+1] = t2.y;
        }
        const float* mk = &ms[k * VV * VV];
#pragma unroll
        for (int u = 0; u < VV; u += 2) {
            v2f s = {0.f, 0.f};
#pragma unroll
            for (int w = 0; w < VV; ++w) {
                v2f m2 = *(const v2f*)(mk + w*VV + u);
                s += xr[w] * m2;             // packed f32 FMA
            }
            zbuf[(ztt*VV + u    )*ZSTR + zc] = s.x;
            zbuf[(ztt*VV + u + 1)*ZSTR + zc] = s.y;
        }
        __syncthreads();

        // GEMM chunk: 16 K-steps of V_WMMA_F32_16X16X4_F32 per tile
#pragma unroll
        for (int ti = 0; ti < 3; ++ti) {
            int tile = wave + 16*ti;
            if (tile < 36) {
                int tm = tile >> 2, tn = tile & 3;
                // A frag: lane -> row M = l16, VGPR pair K = 4s + 2*half + {0,1}
                const float* za = &zbuf[(tm*16 + l16)*ZSTR + 2*half];
                // B frag: lane -> col N = l16, VGPR pair K = 4s + 2*half + {0,1}
                const float* wb = &wfs [(tn*16 + l16)*WSTR + 2*half];
                v8f a8 = acc[ti];
#pragma unroll
                for (int s = 0; s < 16; ++s) {
                    v2f av = *(const v2f*)(za + 4*s);
                    v2f bv = *(const v2f*)(wb + 4*s);
                    a8 = __builtin_amdgcn_wmma_f32_16x16x4_f32(
                        false, av, false, bv, (short)0, a8, false, false);
                }
                acc[ti] = a8;
            }
        }
    }

    __syncthreads();   // all GEMM reads of zbuf complete -> reuse as out-stage

    float* outb = zbuf;  // [o][row], OSTR=145, 9280 <= 9792
#pragma unroll
    for (int ti = 0; ti < 3; ++ti) {
        int tile = wave + 16*ti;
        if (tile < 36) {
            int tm = tile >> 2, tn = tile & 3;
            int o  = tn*16 + l16;
#pragma unroll
            for (int r = 0; r < 8; ++r) {
                int row = tm*16 + half*8 + r;   // C/D layout: M = r + 8*half
                float v = acc[ti][r];
                outb[o*OSTR + row] = v > 0.f ? v : 0.f;  // relu
            }
        }
    }
    __syncthreads();

    // coalesced store: one (o, tt) row of 18 floats per thread
    {
        int o  = tid >> 3;
        int tt = tid & 7;
        const float* src = &outb[o*OSTR + tt*VV];
        float* dst = out + ((size_t)(n*CO + o) * TDIM + (t0 + tt)) * VV;
#pragma unroll
        for (int u = 0; u < VV; u += 2) {
            v2f t2; t2.x = src[u]; t2.y = src[u+1];
            *(v2f*)(dst + u) = t2;
        }
    }
}

extern "C" void kernel_launch(void* const* d_in, const int* in_sizes, int n_in,
                              void* d_out, int out_size, void* d_ws, size_t ws_size,
                              hipStream_t stream) {
    const float* x    = (const float*)d_in[0];
    const float* A    = (const float*)d_in[1];
    const float* PA   = (const float*)d_in[2];
    const float* Wta  = (const float*)d_in[3];
    const float* bta  = (const float*)d_in[4];
    const float* g_ta = (const float*)d_in[5];
    const float* b_ta = (const float*)d_in[6];
    const float* m_ta = (const float*)d_in[7];
    const float* v_ta = (const float*)d_in[8];
    // d_in[9..12] (Wsa/bsa/Wsb/bsb): dead attention branch, output-independent
    const float* g_bn = (const float*)d_in[13];
    const float* b_bn = (const float*)d_in[14];
    const float* m_bn = (const float*)d_in[15];
    const float* v_bn = (const float*)d_in[16];
    float* ws  = (float*)d_ws;
    float* outp = (float*)d_out;

    ctrgc_setup<<<1, 256, 0, stream>>>(A, PA, Wta, bta, g_ta, b_ta, m_ta, v_ta,
                                       g_bn, b_bn, m_bn, v_bn, ws);
    ctrgc_main<<<128 * 32, THREADS, 0, stream>>>(x, ws, outp);
}